// batchGCN_28621662060800
// MI455X (gfx1250) — compile-verified
//
#include <hip/hip_runtime.h>

// Problem constants (match reference)
#define N_NODES   50000
#define N_EDGES   800000
#define BGRAPH    4
#define DIM       64
#define NEG_SLOPE 0.2f

typedef __attribute__((ext_vector_type(2))) float v2f;
typedef __attribute__((ext_vector_type(8))) float v8f;

// ---------------------------------------------------------------------------
// GEMM: out[nrows,64] = act(x[nrows,64]) @ W[64,64]
// One wave32 computes one 16x16 output tile via V_WMMA_F32_16X16X4_F32,
// looping K=64 in 16 steps. W is staged in LDS (16 KB).
//
// Fragment layouts (CDNA5 ISA 7.12.2):
//   A 16x4 (f32):  lanes 0-15 -> M=lane,  VGPR{0,1} = K{0,1}
//                  lanes 16-31 -> M=lane-16, VGPR{0,1} = K{2,3}
//   B 4x16:        mirrored (lane = N, VGPRs = K pair)
//   C/D 16x16:     VGPR v: lanes 0-15 -> M=v, N=lane; lanes 16-31 -> M=v+8
// ---------------------------------------------------------------------------
template <bool APPLY_LRELU>
__global__ void gemm64_wmma(const float* __restrict__ x,
                            const float* __restrict__ W,
                            float* __restrict__ out, int nrows) {
    __shared__ float Wlds[DIM * DIM];
    for (int i = threadIdx.x; i < DIM * DIM; i += blockDim.x) Wlds[i] = W[i];
    __syncthreads();

    const int wave     = blockIdx.x * (blockDim.x >> 5) + (threadIdx.x >> 5);
    const int lane     = threadIdx.x & 31;
    const int tilesM   = nrows >> 4;          // 16-row tiles
    const int tileM    = wave >> 2;           // 4 col-tiles (64/16)
    const int tileN    = wave & 3;
    if (tileM >= tilesM) return;

    const int laneLo = lane & 15;             // M (for A) / N (for B,C)
    const int laneHi = lane >> 4;             // selects K pair {0,1} vs {2,3}

    // A row pointer: row = tileM*16 + laneLo, starting at K offset laneHi*2
    const float* __restrict__ arow = x + ((size_t)(tileM * 16 + laneLo)) * DIM + laneHi * 2;
    const int bcol = tileN * 16 + laneLo;

    v8f acc = {};
#pragma unroll
    for (int k0 = 0; k0 < DIM; k0 += 4) {
        v2f a;
        a.x = arow[k0];
        a.y = arow[k0 + 1];
        if (APPLY_LRELU) {
            a.x = a.x > 0.0f ? a.x : a.x * NEG_SLOPE;
            a.y = a.y > 0.0f ? a.y : a.y * NEG_SLOPE;
        }
        const int kb = k0 + laneHi * 2;
        v2f b;
        b.x = Wlds[kb * DIM + bcol];
        b.y = Wlds[(kb + 1) * DIM + bcol];
        acc = __builtin_amdgcn_wmma_f32_16x16x4_f32(
            /*neg_a=*/false, a, /*neg_b=*/false, b,
            /*c_mod=*/(short)0, acc, /*reuse_a=*/false, /*reuse_b=*/false);
    }

    // Store D: VGPR v -> row (laneHi*8 + v), col (tileN*16 + laneLo)
    float* __restrict__ orow =
        out + ((size_t)(tileM * 16 + laneHi * 8)) * DIM + tileN * 16 + laneLo;
#pragma unroll
    for (int v = 0; v < 8; ++v) orow[(size_t)v * DIM] = acc[v];
}

// ---------------------------------------------------------------------------
// dst[i*64 + j] = bias[j]  (broadcast-init so scatter needs no epilogue)
// ---------------------------------------------------------------------------
__global__ void init_bias(float* __restrict__ dst, const float* __restrict__ bias) {
    const int gid = blockIdx.x * blockDim.x + threadIdx.x;  // grid covers N*64 exactly
    dst[gid] = bias[gid & (DIM - 1)];
}

// ---------------------------------------------------------------------------
// Edge aggregation: dst[rows[e]] += ew[e] * sup[cols[e]]
// 16 lanes per edge; each lane handles 4 features via a float4 gather and
// 4 global_atomic_add_f32. Gather + scatter working sets are L2-resident.
// ---------------------------------------------------------------------------
__global__ void scatter_edges(const long long* __restrict__ rows,
                              const long long* __restrict__ cols,
                              const float* __restrict__ ew,
                              const float* __restrict__ sup,
                              float* __restrict__ dst) {
    const int gid = blockIdx.x * blockDim.x + threadIdx.x;
    const int e = gid >> 4;          // edge id
    const int q = gid & 15;          // feature quad (4 floats each)
    if (e >= N_EDGES) return;

    const int r = (int)rows[e];
    const int c = (int)cols[e];
    const float w = ew[e];

    const float4 s = *(const float4*)(sup + (size_t)c * DIM + q * 4);
    float* __restrict__ p = dst + (size_t)r * DIM + q * 4;
    atomicAdd(p + 0, w * s.x);
    atomicAdd(p + 1, w * s.y);
    atomicAdd(p + 2, w * s.z);
    atomicAdd(p + 3, w * s.w);
}

// ---------------------------------------------------------------------------
// Inputs (reference dict order):
//  0: bx [N,64] f32         1: edge_index [B,2,E] i64   2: edge_weight [B,E] f32
//  3: W0 [64,64] f32        4: b0 [64] f32
//  5: W1 [64,64] f32        6: b1 [64] f32
// Output: [B,N,64] f32
// ---------------------------------------------------------------------------
extern "C" void kernel_launch(void* const* d_in, const int* in_sizes, int n_in,
                              void* d_out, int out_size, void* d_ws, size_t ws_size,
                              hipStream_t stream) {
    const float*     bx = (const float*)d_in[0];
    const long long* ei = (const long long*)d_in[1];
    const float*     ew = (const float*)d_in[2];
    const float*     W0 = (const float*)d_in[3];
    const float*     b0 = (const float*)d_in[4];
    const float*     W1 = (const float*)d_in[5];
    const float*     b1 = (const float*)d_in[6];
    float* out = (float*)d_out;

    float* ws       = (float*)d_ws;
    float* support0 = ws;                               // [N,64]
    float* x1       = ws + (size_t)N_NODES * DIM;       // [N,64]
    float* support1 = ws + (size_t)2 * N_NODES * DIM;   // [N,64]

    const dim3 blk(256);
    const int wavesTotal    = (N_NODES / 16) * (DIM / 16);     // 12500 tiles
    const int gemmBlocks    = (wavesTotal + 7) / 8;            // 8 waves/block
    const int initBlocks    = (N_NODES * DIM) / 256;           // exact
    const int scatterBlocks = (N_EDGES * 16) / 256;            // exact

    // Layer-0 dense part is graph-independent: compute once.
    gemm64_wmma<false><<<gemmBlocks, blk, 0, stream>>>(bx, W0, support0, N_NODES);

    for (int b = 0; b < BGRAPH; ++b) {
        const long long* rows = ei + (size_t)b * 2 * N_EDGES;
        const long long* cols = rows + N_EDGES;
        const float*     ewb  = ew + (size_t)b * N_EDGES;
        float*           outb = out + (size_t)b * N_NODES * DIM;

        // Layer 0 aggregation: x1 = b0 + segsum(ew * support0[cols])
        init_bias<<<initBlocks, blk, 0, stream>>>(x1, b0);
        scatter_edges<<<scatterBlocks, blk, 0, stream>>>(rows, cols, ewb, support0, x1);

        // Layer 1 dense part with fused leaky_relu on input: support1 = lrelu(x1) @ W1
        gemm64_wmma<true><<<gemmBlocks, blk, 0, stream>>>(x1, W1, support1, N_NODES);

        // Layer 1 aggregation straight into the output slice.
        init_bias<<<initBlocks, blk, 0, stream>>>(outb, b1);
        scatter_edges<<<scatterBlocks, blk, 0, stream>>>(rows, cols, ewb, support1, outb);
    }
}